// DiffusionConv_60533269069994
// MI455X (gfx1250) — compile-verified
//
#include <hip/hip_runtime.h>
#include <hip/hip_bf16.h>

// ---------------------------------------------------------------------------
// DiffusionConv for MI455X (gfx1250, wave32)
//   phase 1: 3x edge scatter-add diffusion (global_atomic_add_f32, L2-resident)
//   phase 2: (200000 x 224) @ (224 x 64) + bias via V_WMMA_F32_16X16X4_F32
// ---------------------------------------------------------------------------

typedef __attribute__((ext_vector_type(2))) float v2f;
typedef __attribute__((ext_vector_type(8))) float v8f;

constexpr int   kN     = 25000;
constexpr int   kT     = 8;
constexpr int   kInF   = 32;
constexpr int   kOutF  = 64;
constexpr int   kSteps = 3;
constexpr int   kKTot  = (2 * kSteps + 1) * kInF;   // 224
constexpr int   kE     = 800000;
constexpr int   kRow   = kT * kInF;                 // 256 floats per node
constexpr int   kNT    = kN * kT;                   // 200000 output rows
constexpr int   kHalf  = kNT * kInF;                // 6,400,000 floats (one half-state)
constexpr int   kState = 2 * kHalf;                 // 12,800,000 floats per state buffer

// ---------------- element-wise init kernels --------------------------------

__global__ void k_zero(float4* __restrict__ p) {
    int i = blockIdx.x * blockDim.x + threadIdx.x;     // grid sized exactly
    p[i] = make_float4(0.f, 0.f, 0.f, 0.f);
}

// out = -concat(feat, feat)   (i.e. -x0)
__global__ void k_negdup(const float4* __restrict__ feat, float4* __restrict__ out) {
    int i = blockIdx.x * blockDim.x + threadIdx.x;     // 0 .. kState/4
    int half4 = kHalf / 4;
    float4 v = feat[i < half4 ? i : i - half4];
    out[i] = make_float4(-v.x, -v.y, -v.z, -v.w);
}

// out = -in
__global__ void k_neg(const float4* __restrict__ in, float4* __restrict__ out) {
    int i = blockIdx.x * blockDim.x + threadIdx.x;
    float4 v = in[i];
    out[i] = make_float4(-v.x, -v.y, -v.z, -v.w);
}

// ---------------- edge scatter-add (segment_sum) ---------------------------
// One thread per (edge, 4-float chunk): E * 64 threads. in_lo serves src<N,
// in_hi serves src>=N (for feat both are the same pointer -> implicit concat).

__global__ void k_scatter(const int*   __restrict__ src,
                          const int*   __restrict__ dst,
                          const float* __restrict__ ef,
                          const float* __restrict__ in_lo,
                          const float* __restrict__ in_hi,
                          float*       __restrict__ out,
                          float scale) {
    int idx = blockIdx.x * blockDim.x + threadIdx.x;   // grid sized exactly E*64
    int e = idx >> 6;
    int c = (idx & 63) << 2;
    int s = src[e];
    const float* base = (s < kN) ? (in_lo + (size_t)s * kRow)
                                 : (in_hi + (size_t)(s - kN) * kRow);
    float w = ef[e] * scale;
    float4 v = *(const float4*)(base + c);
    float* o = out + (size_t)dst[e] * kRow + c;
    atomicAdd(o + 0, v.x * w);
    atomicAdd(o + 1, v.y * w);
    atomicAdd(o + 2, v.z * w);
    atomicAdd(o + 3, v.w * w);
}

// ---------------- WMMA GEMM: h(200000 x 224) @ W(224 x 64) + b -------------
// 128 threads = 4 waves per block; each wave owns one 16-row M tile and all
// four 16-col N tiles (4 x v8f accumulators). K walked in 4-wide f32 WMMA
// steps across the 7 concatenated 32-feature source blocks. Each block's base
// pointer is passed straight through from a kernel argument so the A-stream
// lowers to global_load_b64 (LOADcnt only) instead of flat_load_b64, keeping
// it out of the DScnt domain that the LDS B-fragment reads live in.

__device__ __forceinline__ void gemm_block(const float* __restrict__ base,
                                           int kgBase, int Mm, int kh2, int m,
                                           const float* __restrict__ ldsW,
                                           v8f acc[4]) {
    const float* A = base + (size_t)Mm * kInF;
#pragma unroll
    for (int k4 = 0; k4 < 8; ++k4) {
        const int k = k4 * 4 + kh2;           // per-lane K offset within block
        // A fragment 16x4 f32: lane<16 holds (M=lane, K=k..k+1),
        //                      lane>=16 holds (M=lane-16, K=k+2..k+3)
        v2f a = { A[k], A[k + 1] };           // 8B contiguous global load
        const int kg = kgBase + k;
        const float* Wr0 = &ldsW[(size_t)kg * kOutF];
        const float* Wr1 = &ldsW[(size_t)(kg + 1) * kOutF];
#pragma unroll
        for (int j = 0; j < 4; ++j) {
            const int n = j * 16 + m;
            v2f bb = { Wr0[n], Wr1[n] };      // B fragment 4x16 f32 (mirrored)
            acc[j] = __builtin_amdgcn_wmma_f32_16x16x4_f32(
                /*neg_a=*/false, a, /*neg_b=*/false, bb,
                /*c_mod=*/(short)0, acc[j],
                /*reuse_a=*/false, /*reuse_b=*/false);
        }
    }
}

__global__ __launch_bounds__(128) void k_gemm(const float* __restrict__ feat,
                                              const float* __restrict__ s1,
                                              const float* __restrict__ s2,
                                              const float* __restrict__ s3,
                                              const float* __restrict__ W,
                                              const float* __restrict__ bias,
                                              float*       __restrict__ out) {
    __shared__ float ldsW[kKTot * kOutF];   // 57344 B of the 320 KB WGP LDS
    __shared__ float ldsB[kOutF];

    for (int i = threadIdx.x; i < kKTot * kOutF; i += blockDim.x) ldsW[i] = W[i];
    if (threadIdx.x < kOutF) ldsB[threadIdx.x] = bias[threadIdx.x];
    __syncthreads();

    const int wave  = threadIdx.x >> 5;
    const int lane  = threadIdx.x & 31;
    const int tile  = blockIdx.x * 4 + wave;      // 12500 tiles exactly
    const int Mbase = tile * 16;
    const int m     = lane & 15;                  // M within tile for A / N for B,D
    const int kh2   = (lane >> 4) * 2;            // K sub-offset: lanes 16-31 -> K+2,K+3
    const int Mm    = Mbase + m;

    v8f zero = {0.f, 0.f, 0.f, 0.f, 0.f, 0.f, 0.f, 0.f};
    v8f acc[4] = {zero, zero, zero, zero};

    // 7 concatenated 32-feature blocks: feat, s1[:n], s1[n:], s2[:n], s2[n:], s3[:n], s3[n:]
    gemm_block(feat,                     0 * 32, Mm, kh2, m, ldsW, acc);
    gemm_block(s1,                       1 * 32, Mm, kh2, m, ldsW, acc);
    gemm_block(s1 + (size_t)kHalf,       2 * 32, Mm, kh2, m, ldsW, acc);
    gemm_block(s2,                       3 * 32, Mm, kh2, m, ldsW, acc);
    gemm_block(s2 + (size_t)kHalf,       4 * 32, Mm, kh2, m, ldsW, acc);
    gemm_block(s3,                       5 * 32, Mm, kh2, m, ldsW, acc);
    gemm_block(s3 + (size_t)kHalf,       6 * 32, Mm, kh2, m, ldsW, acc);

    // D layout: VGPR vr -> M = Mbase + vr (+8 for lanes 16-31), N = j*16 + (lane&15)
    // Output is written once and never re-read this dispatch -> NT stores so the
    // 51 MB stream does not evict the L2-resident diffusion states.
    const int rowOff = (lane >> 4) * 8;
#pragma unroll
    for (int j = 0; j < 4; ++j) {
        const int n = j * 16 + m;
        const float bv = ldsB[n];
#pragma unroll
        for (int vr = 0; vr < 8; ++vr) {
            const int row = Mbase + rowOff + vr;
            __builtin_nontemporal_store(acc[j][vr] + bv, &out[(size_t)row * kOutF + n]);
        }
    }
}

// ---------------------------------------------------------------------------

extern "C" void kernel_launch(void* const* d_in, const int* in_sizes, int n_in,
                              void* d_out, int out_size, void* d_ws, size_t ws_size,
                              hipStream_t stream) {
    const float* feat = (const float*)d_in[0];   // (N, T, 32)
    const float* ef   = (const float*)d_in[1];   // (E,)
    const float* W    = (const float*)d_in[2];   // (224, 64)
    const float* bias = (const float*)d_in[3];   // (64,)
    const int*   src  = (const int*)d_in[4];     // (E,)
    const int*   dst  = (const int*)d_in[5];     // (E,)
    float*       out  = (float*)d_out;           // (N, T, 64)

    float* s1 = (float*)d_ws;                    // each state: 2N*T*32 f32 = 51.2 MB
    float* s2 = s1 + (size_t)kState;
    float* s3 = s2 + (size_t)kState;

    constexpr int BT = 256;
    constexpr int gState4  = kState / 4 / BT;    // 12500 blocks (exact)
    constexpr int gScatter = (kE * 64) / BT;     // 200000 blocks (exact)

    // x1 = spmm(x0)
    k_zero   <<<gState4,  BT, 0, stream>>>((float4*)s1);
    k_scatter<<<gScatter, BT, 0, stream>>>(src, dst, ef, feat, feat, s1, 1.0f);
    // x2 = 2*spmm(x1) - x0   (init s2 = -x0, scatter with weight 2*ef)
    k_negdup <<<gState4,  BT, 0, stream>>>((const float4*)feat, (float4*)s2);
    k_scatter<<<gScatter, BT, 0, stream>>>(src, dst, ef, s1, s1 + (size_t)kHalf, s2, 2.0f);
    // x3 = 2*spmm(x2) - x1   (init s3 = -x1)
    k_neg    <<<gState4,  BT, 0, stream>>>((const float4*)s1, (float4*)s3);
    k_scatter<<<gScatter, BT, 0, stream>>>(src, dst, ef, s2, s2 + (size_t)kHalf, s3, 2.0f);
    // out = concat([feat, s1lo, s1hi, s2lo, s2hi, s3lo, s3hi], -1) @ W + b
    k_gemm   <<<(kNT / 16) / 4, 128, 0, stream>>>(feat, s1, s2, s3, W, bias, out);
}